// DCNv3_pytorch_31061203485452
// MI455X (gfx1250) — compile-verified
//
#include <hip/hip_runtime.h>
#include <math.h>

// ---------------------------------------------------------------------------
// DCNv3 block for MI455X (gfx1250, wave32).
// Pipeline: in_proj GEMM (fp32 WMMA) -> dwconv+LN+GELU -> offset/mask GEMMs
//           -> softmax(P=9) -> bilinear gather core -> out_proj GEMM (NCHW).
// ---------------------------------------------------------------------------

typedef __attribute__((ext_vector_type(2))) float v2f;
typedef __attribute__((ext_vector_type(8))) float v8f;

#define CDIM 512
#define HH 64
#define WW 64
#define HW 4096           // HH*WW
#define NIMG 2
#define MROWS (NIMG * HW) // 8192 pixels
#define GRP 16
#define CG 32
#define PP 9

#define MBLK 128
#define NBLK 64
#define KBLK 32
#define KPAD 34           // even row stride: b64-aligned fragments, no bank conflicts

// ---------------------------------------------------------------------------
// fp32 WMMA GEMM: C[M x Nout] = A[M x K] * B[K x Nout] + bias
//   amode  : 0 = A row-major (pixel,K); 1 = A is NCHW (gather A[(n*K+k)*HW+hw])
//   epmode : 0 = row-major store; 1 = store transposed to NCHW (Nout==CDIM)
// Block: 256 threads = 8 waves stacked in M; wave tile 16(M) x 64(N) = 4 accs.
// B tile staged TRANSPOSED in LDS so every WMMA operand is one ds_load_b64.
// ---------------------------------------------------------------------------
__global__ __launch_bounds__(256) void gemm_wmma_f32(
    const float* __restrict__ A, const float* __restrict__ B,
    const float* __restrict__ bias, float* __restrict__ C,
    int Nout, int Kdim, int amode, int epmode)
{
    __shared__ __align__(16) float As[MBLK][KPAD];  // 128 x 34  (~17.4 KB)
    __shared__ __align__(16) float Bt[NBLK][KPAD];  //  64 x 34  (~8.7 KB)

    const int tid   = threadIdx.x;
    const int wave  = tid >> 5;
    const int lane  = tid & 31;
    const int l16   = lane & 15;
    const int lhalf = lane >> 4;              // 0: K={0,1}, 1: K={2,3}
    const int m0 = blockIdx.x * MBLK;
    const int n0 = blockIdx.y * NBLK;
    const int wm = wave << 4;                 // 0,16,...,112

    v8f acc[4] = {{}, {}, {}, {}};

    for (int k0 = 0; k0 < Kdim; k0 += KBLK) {
        if (amode == 0) {
            for (int i = tid; i < MBLK * KBLK; i += 256) {
                int r = i >> 5, c = i & (KBLK - 1);  // contiguous k per row
                As[r][c] = A[(size_t)(m0 + r) * Kdim + (k0 + c)];
            }
        } else {
            for (int i = tid; i < MBLK * KBLK; i += 256) {
                int r = i & (MBLK - 1), c = i >> 7;  // contiguous hw per k
                int row = m0 + r;
                int n = row >> 12, hw = row & (HW - 1);
                As[r][c] = A[((size_t)n * Kdim + (k0 + c)) * HW + hw];
            }
        }
        // B tile loaded coalesced along n, stored transposed: Bt[n][k]
        for (int i = tid; i < KBLK * NBLK; i += 256) {
            int r = i >> 6, c = i & (NBLK - 1);
            int nn = n0 + c;
            Bt[c][r] = (nn < Nout) ? B[(size_t)(k0 + r) * Nout + nn] : 0.0f;
        }
        __syncthreads();

#pragma unroll
        for (int kk = 0; kk < KBLK; kk += 4) {
            const int ka = kk + (lhalf << 1);
            const v2f a  = *(const v2f*)(&As[wm + l16][ka]);
            const v2f b0 = *(const v2f*)(&Bt[ 0 + l16][ka]);
            const v2f b1 = *(const v2f*)(&Bt[16 + l16][ka]);
            const v2f b2 = *(const v2f*)(&Bt[32 + l16][ka]);
            const v2f b3 = *(const v2f*)(&Bt[48 + l16][ka]);
            acc[0] = __builtin_amdgcn_wmma_f32_16x16x4_f32(
                false, a, false, b0, (short)0, acc[0], false, false);
            acc[1] = __builtin_amdgcn_wmma_f32_16x16x4_f32(
                false, a, false, b1, (short)0, acc[1], false, false);
            acc[2] = __builtin_amdgcn_wmma_f32_16x16x4_f32(
                false, a, false, b2, (short)0, acc[2], false, false);
            acc[3] = __builtin_amdgcn_wmma_f32_16x16x4_f32(
                false, a, false, b3, (short)0, acc[3], false, false);
        }
        __syncthreads();
    }

    // D layout: lanes 0-15 -> N=l16, VGPR r -> M=r; lanes 16-31 -> M=r+8.
#pragma unroll
    for (int t = 0; t < 4; ++t) {
        const int col = n0 + t * 16 + l16;
        if (col < Nout) {
            const float bv = bias[col];
#pragma unroll
            for (int r = 0; r < 8; ++r) {
                const int row = m0 + wm + (lhalf << 3) + r;
                const float v = acc[t][r] + bv;
                if (epmode == 0) {
                    C[(size_t)row * Nout + col] = v;
                } else {
                    const int n = row >> 12, hw = row & (HW - 1);
                    C[((size_t)n * Nout + col) * HW + hw] = v;
                }
            }
        }
    }
}

// ---------------------------------------------------------------------------
// Depthwise 3x3 (zero pad) + bias, LayerNorm over C, exact GELU.
// One block (128 thr x 4 ch) per pixel; x is NCHW, x1 out is NHWC.
// ---------------------------------------------------------------------------
__global__ __launch_bounds__(128) void dwconv_ln_gelu(
    const float* __restrict__ x, const float* __restrict__ dw_w,
    const float* __restrict__ dw_b, const float* __restrict__ ln_g,
    const float* __restrict__ ln_b, float* __restrict__ x1)
{
    const int pix = blockIdx.x;
    const int n = pix >> 12, hw = pix & (HW - 1);
    const int h = hw >> 6, w = hw & (WW - 1);
    const int tid = threadIdx.x;

    const float* xb = x + (size_t)n * CDIM * HW;
    float v[4];
#pragma unroll
    for (int j = 0; j < 4; ++j) {
        const int c = tid * 4 + j;
        const float* xc = xb + (size_t)c * HW;
        const float* wc = dw_w + c * 9;
        float acc = dw_b[c];
#pragma unroll
        for (int dy = 0; dy < 3; ++dy) {
            const int yy = h + dy - 1;
            if (yy < 0 || yy >= HH) continue;
#pragma unroll
            for (int dx = 0; dx < 3; ++dx) {
                const int xx = w + dx - 1;
                if (xx < 0 || xx >= WW) continue;
                acc += wc[dy * 3 + dx] * xc[yy * WW + xx];
            }
        }
        v[j] = acc;
    }

    __shared__ float ssum[128], ssq[128];
    float s = v[0] + v[1] + v[2] + v[3];
    float q = v[0] * v[0] + v[1] * v[1] + v[2] * v[2] + v[3] * v[3];
    ssum[tid] = s;
    ssq[tid] = q;
    __syncthreads();
    for (int off = 64; off > 0; off >>= 1) {
        if (tid < off) {
            ssum[tid] += ssum[tid + off];
            ssq[tid]  += ssq[tid + off];
        }
        __syncthreads();
    }
    const float mu  = ssum[0] * (1.0f / (float)CDIM);
    const float var = ssq[0] * (1.0f / (float)CDIM) - mu * mu;
    const float rs  = rsqrtf(var + 1e-5f);

    float* out = x1 + (size_t)pix * CDIM;
#pragma unroll
    for (int j = 0; j < 4; ++j) {
        const int c = tid * 4 + j;
        const float y = (v[j] - mu) * rs * ln_g[c] + ln_b[c];
        out[c] = 0.5f * y * (1.0f + erff(y * 0.70710678118654752f));
    }
}

// ---------------------------------------------------------------------------
// Softmax over P=9 per (pixel, group), in place. 9 contiguous floats/thread.
// ---------------------------------------------------------------------------
__global__ __launch_bounds__(256) void softmax_p9(float* __restrict__ m)
{
    const int idx = blockIdx.x * blockDim.x + threadIdx.x;
    if (idx >= MROWS * GRP) return;
    float* p = m + (size_t)idx * PP;
    float e[PP];
    float mx = p[0];
#pragma unroll
    for (int i = 1; i < PP; ++i) mx = fmaxf(mx, p[i]);
    float s = 0.0f;
#pragma unroll
    for (int i = 0; i < PP; ++i) { e[i] = expf(p[i] - mx); s += e[i]; }
    const float inv = 1.0f / s;
#pragma unroll
    for (int i = 0; i < PP; ++i) p[i] = e[i] * inv;
}

// ---------------------------------------------------------------------------
// DCNv3 bilinear-gather core. One wave per (pixel, group); lane = channel.
// Padded-frame sample position simplifies to: px = w + kx + ox, py = h + ky + oy
// with taps zero outside the unpadded image.
// ---------------------------------------------------------------------------
__global__ __launch_bounds__(256) void dcn_core(
    const float* __restrict__ xproj,  // (pixel, C) NHWC
    const float* __restrict__ off,    // (pixel, G*P*2)
    const float* __restrict__ msk,    // (pixel, G*P) softmaxed
    float* __restrict__ out)          // (pixel, C)
{
    const int wid  = blockIdx.x * 8 + (threadIdx.x >> 5);
    const int lane = threadIdx.x & 31;
    const int g   = wid & (GRP - 1);
    const int pix = wid >> 4;
    const int n = pix >> 12, hw = pix & (HW - 1);
    const int h = hw >> 6, w = hw & (WW - 1);
    const int c = g * CG + lane;

    const float* img = xproj + (size_t)n * HW * CDIM;
    const float* op = off + (size_t)pix * (GRP * PP * 2) + g * (PP * 2);
    const float* mp = msk + (size_t)pix * (GRP * PP) + g * PP;

    auto tap = [&](int yy, int xx) -> float {
        if (yy < 0 || yy >= HH || xx < 0 || xx >= WW) return 0.0f;
        return img[((size_t)yy * WW + xx) * CDIM + c];
    };

    float acc = 0.0f;
#pragma unroll
    for (int p = 0; p < PP; ++p) {
        const int ki = p / 3;         // x varies slowest (torch ordering)
        const int kj = p - ki * 3;
        const float ox = op[p * 2 + 0];
        const float oy = op[p * 2 + 1];
        const float fx = (float)(w + ki - 1) + ox;   // unpadded-frame coords
        const float fy = (float)(h + kj - 1) + oy;
        const float x0f = floorf(fx), y0f = floorf(fy);
        const int x0 = (int)x0f, y0 = (int)y0f;
        const float wx = fx - x0f, wy = fy - y0f;
        const float v00 = tap(y0, x0),     v01 = tap(y0, x0 + 1);
        const float v10 = tap(y0 + 1, x0), v11 = tap(y0 + 1, x0 + 1);
        const float sv = (v00 * (1.0f - wx) + v01 * wx) * (1.0f - wy)
                       + (v10 * (1.0f - wx) + v11 * wx) * wy;
        acc += mp[p] * sv;
    }
    out[(size_t)pix * CDIM + c] = acc;
}

// ---------------------------------------------------------------------------
extern "C" void kernel_launch(void* const* d_in, const int* in_sizes, int n_in,
                              void* d_out, int out_size, void* d_ws, size_t ws_size,
                              hipStream_t stream)
{
    (void)in_sizes; (void)n_in; (void)out_size; (void)ws_size;

    const float* x      = (const float*)d_in[0];
    const float* dw_w   = (const float*)d_in[1];
    const float* dw_b   = (const float*)d_in[2];
    const float* ln_g   = (const float*)d_in[3];
    const float* ln_b   = (const float*)d_in[4];
    const float* off_w  = (const float*)d_in[5];
    const float* off_b  = (const float*)d_in[6];
    const float* mask_w = (const float*)d_in[7];
    const float* mask_b = (const float*)d_in[8];
    const float* in_w   = (const float*)d_in[9];
    const float* in_b   = (const float*)d_in[10];
    const float* out_w  = (const float*)d_in[11];
    const float* out_b  = (const float*)d_in[12];
    float* out = (float*)d_out;

    float* ws    = (float*)d_ws;
    float* xproj = ws;                               // 8192*512
    float* x1    = xproj + (size_t)MROWS * CDIM;     // 8192*512
    float* offl  = x1    + (size_t)MROWS * CDIM;     // 8192*288
    float* mskl  = offl  + (size_t)MROWS * (GRP * PP * 2);  // 8192*144
    float* core  = mskl  + (size_t)MROWS * (GRP * PP);      // 8192*512

    const dim3 blk(256);

    // 1. input_proj: xproj = NHWC(x) @ in_w + in_b
    gemm_wmma_f32<<<dim3(MROWS / MBLK, CDIM / NBLK), blk, 0, stream>>>(
        x, in_w, in_b, xproj, CDIM, CDIM, /*amode=*/1, /*epmode=*/0);

    // 2. depthwise conv + LayerNorm + GELU -> x1 (NHWC)
    dwconv_ln_gelu<<<dim3(MROWS), dim3(128), 0, stream>>>(
        x, dw_w, dw_b, ln_g, ln_b, x1);

    // 3. offsets = x1 @ off_w + off_b  (Nout = 288)
    gemm_wmma_f32<<<dim3(MROWS / MBLK, (GRP * PP * 2 + NBLK - 1) / NBLK), blk, 0, stream>>>(
        x1, off_w, off_b, offl, GRP * PP * 2, CDIM, 0, 0);

    // 4. mask logits = x1 @ mask_w + mask_b  (Nout = 144)
    gemm_wmma_f32<<<dim3(MROWS / MBLK, (GRP * PP + NBLK - 1) / NBLK), blk, 0, stream>>>(
        x1, mask_w, mask_b, mskl, GRP * PP, CDIM, 0, 0);

    // 5. softmax over P=9 (in place)
    softmax_p9<<<dim3((MROWS * GRP + 255) / 256), blk, 0, stream>>>(mskl);

    // 6. deformable gather core
    dcn_core<<<dim3(MROWS * GRP / 8), blk, 0, stream>>>(xproj, offl, mskl, core);

    // 7. output_proj with NCHW transpose epilogue
    gemm_wmma_f32<<<dim3(MROWS / MBLK, CDIM / NBLK), blk, 0, stream>>>(
        core, out_w, out_b, out, CDIM, CDIM, 0, /*epmode=*/1);
}